// AWB_23175643529520
// MI455X (gfx1250) — compile-verified
//
#include <hip/hip_runtime.h>
#include <hip/hip_bf16.h>

#define NUM_CLASSES 100
#define TILE_ROWS   256
#define BLOCK       256
#define ROW_BYTES   (NUM_CLASSES * 4)            // 400 B per row
#define TILE_BYTES  (TILE_ROWS * ROW_BYTES)      // 102400 B per tile
#define COPY_CHUNKS (TILE_BYTES / (BLOCK * 16))  // 25 async b128 ops per thread per tile
#define BINS        (4 * NUM_CLASSES)            // count, sum(-log), sum(pt), sum(pt^2)

static_assert(BLOCK == 256 && TILE_BYTES % (BLOCK * 16) == 0, "chunk layout");
static_assert(COPY_CHUNKS == 25, "clause + s_wait_asynccnt immediates assume 25 ops/tile");

// Low 32 bits of a generic pointer into LDS == wave-relative LDS byte offset.
__device__ __forceinline__ unsigned lds_byte_off(const void* p) {
  return (unsigned)(unsigned long long)p;
}

// CDNA5 async global->LDS streaming copy. IOFFSET is added to BOTH the LDS
// destination (vdst + off) and the global source (vaddr + off), so a single
// LDS-offset VGPR + one 64-bit address pair covers the whole 100KB tile.
// One s_clause-headed burst of 25 b128 ops per wave => uninterrupted issue,
// ~100KB in flight per WGP (> bandwidth-delay product), zero loop overhead.
__device__ __forceinline__ void async_tile_copy(char* ldsDst, const char* gSrc, int tid) {
  const unsigned l = lds_byte_off(ldsDst) + (unsigned)(tid * 16);
  const char*    g = gSrc + tid * 16;
  asm volatile(
      "s_clause 0x18\n\t"
      "global_load_async_to_lds_b128 %0, %1, off\n\t"
      "global_load_async_to_lds_b128 %0, %1, off offset:4096\n\t"
      "global_load_async_to_lds_b128 %0, %1, off offset:8192\n\t"
      "global_load_async_to_lds_b128 %0, %1, off offset:12288\n\t"
      "global_load_async_to_lds_b128 %0, %1, off offset:16384\n\t"
      "global_load_async_to_lds_b128 %0, %1, off offset:20480\n\t"
      "global_load_async_to_lds_b128 %0, %1, off offset:24576\n\t"
      "global_load_async_to_lds_b128 %0, %1, off offset:28672\n\t"
      "global_load_async_to_lds_b128 %0, %1, off offset:32768\n\t"
      "global_load_async_to_lds_b128 %0, %1, off offset:36864\n\t"
      "global_load_async_to_lds_b128 %0, %1, off offset:40960\n\t"
      "global_load_async_to_lds_b128 %0, %1, off offset:45056\n\t"
      "global_load_async_to_lds_b128 %0, %1, off offset:49152\n\t"
      "global_load_async_to_lds_b128 %0, %1, off offset:53248\n\t"
      "global_load_async_to_lds_b128 %0, %1, off offset:57344\n\t"
      "global_load_async_to_lds_b128 %0, %1, off offset:61440\n\t"
      "global_load_async_to_lds_b128 %0, %1, off offset:65536\n\t"
      "global_load_async_to_lds_b128 %0, %1, off offset:69632\n\t"
      "global_load_async_to_lds_b128 %0, %1, off offset:73728\n\t"
      "global_load_async_to_lds_b128 %0, %1, off offset:77824\n\t"
      "global_load_async_to_lds_b128 %0, %1, off offset:81920\n\t"
      "global_load_async_to_lds_b128 %0, %1, off offset:86016\n\t"
      "global_load_async_to_lds_b128 %0, %1, off offset:90112\n\t"
      "global_load_async_to_lds_b128 %0, %1, off offset:94208\n\t"
      "global_load_async_to_lds_b128 %0, %1, off offset:98304"
      :: "v"(l), "v"(g) : "memory");
}

__device__ __forceinline__ void wait_async_all()  { asm volatile("s_wait_asynccnt 0"  ::: "memory"); }
__device__ __forceinline__ void wait_async_tile() { asm volatile("s_wait_asynccnt 25" ::: "memory"); }

__global__ void __launch_bounds__(BLOCK)
awb_main(const float* __restrict__ logits, const int* __restrict__ target,
         float* __restrict__ gbins, int numTiles) {
  extern __shared__ char smem[];
  float* bins  = reinterpret_cast<float*>(smem);       // 400 floats of class bins
  char*  bufA  = smem + BINS * sizeof(float);          // double-buffered 100KB tiles
  char*  bufB  = bufA + TILE_BYTES;
  const int tid = threadIdx.x;

  for (int i = tid; i < BINS; i += BLOCK) bins[i] = 0.0f;

  const int first = blockIdx.x;
  const int step  = gridDim.x;
  if (first < numTiles)
    async_tile_copy(bufA, (const char*)logits + (size_t)first * TILE_BYTES, tid);

  int cur = 0;
  for (int t = first; t < numTiles; t += step) {
    char* curBuf = cur ? bufB : bufA;
    char* nxtBuf = cur ? bufA : bufB;
    const int nxt = t + step;
    if (nxt < numTiles) {
      async_tile_copy(nxtBuf, (const char*)logits + (size_t)nxt * TILE_BYTES, tid);
      wait_async_tile();      // in-order: first 25 (current tile) done; prefetch in flight
    } else {
      wait_async_all();
    }
    __syncthreads();          // all waves' current-tile data visible in LDS

    // One row per thread out of LDS into registers (25 x ds_load_b128).
    const float* row = (const float*)(curBuf + tid * ROW_BYTES);
    float4 v[NUM_CLASSES / 4];
#pragma unroll
    for (int k = 0; k < NUM_CLASSES / 4; ++k) v[k] = ((const float4*)row)[k];

    float m = v[0].x;
#pragma unroll
    for (int k = 0; k < NUM_CLASSES / 4; ++k) {
      m = fmaxf(m, v[k].x); m = fmaxf(m, v[k].y);
      m = fmaxf(m, v[k].z); m = fmaxf(m, v[k].w);
    }
    float s = 0.0f;
#pragma unroll
    for (int k = 0; k < NUM_CLASSES / 4; ++k) {
      s += __expf(v[k].x - m); s += __expf(v[k].y - m);
      s += __expf(v[k].z - m); s += __expf(v[k].w - m);
    }

    const int   cls = target[(size_t)t * TILE_ROWS + tid];
    const float xt  = row[cls];                 // dynamic index -> ds_load_b32
    const float pt  = __expf(xt - m) / s;       // softmax prob of target class
    const float neg = -__logf(pt + 1e-6f);

    atomicAdd(&bins[cls],                   1.0f);
    atomicAdd(&bins[NUM_CLASSES     + cls], neg);
    atomicAdd(&bins[2 * NUM_CLASSES + cls], pt);
    atomicAdd(&bins[3 * NUM_CLASSES + cls], pt * pt);

    __syncthreads();          // protect curBuf before it becomes the prefetch dst
    cur ^= 1;
  }

  __syncthreads();
  for (int i = tid; i < BINS; i += BLOCK) atomicAdd(&gbins[i], bins[i]);
}

__device__ __forceinline__ float wave_max(float v) {
#pragma unroll
  for (int m = 16; m > 0; m >>= 1) v = fmaxf(v, __shfl_xor(v, m, 32));
  return v;
}
__device__ __forceinline__ float wave_sum(float v) {
#pragma unroll
  for (int m = 16; m > 0; m >>= 1) v += __shfl_xor(v, m, 32);
  return v;
}

// One wave32 folds the 100-class statistics into the scalar loss.
__global__ void awb_final(const float* __restrict__ gbins, float* __restrict__ out, float invN) {
  const int lane = threadIdx.x;
  const float* cnt  = gbins;
  const float* sNeg = gbins + NUM_CLASSES;
  const float* sPt  = gbins + 2 * NUM_CLASSES;
  const float* sPt2 = gbins + 3 * NUM_CLASSES;

  float cmax = 0.0f;
  for (int c = lane; c < NUM_CLASSES; c += 32) cmax = fmaxf(cmax, cnt[c]);
  cmax = wave_max(cmax);

  float amax = -3.4e38f;
  for (int c = lane; c < NUM_CLASSES; c += 32) {
    const float n = cnt[c];
    const float A = (n > 0.0f) ? (__logf(cmax / n) + 1.0f) : 0.0f;
    amax = fmaxf(amax, A);
  }
  amax = wave_max(amax);

  float asum = 0.0f, l1 = 0.0f, l2raw = 0.0f;
  for (int c = lane; c < NUM_CLASSES; c += 32) {
    const float n = cnt[c];
    const float A = (n > 0.0f) ? (__logf(cmax / n) + 1.0f) : 0.0f;
    const float e = __expf(A - amax);   // softmax(Alpha) numerator
    asum += e;
    if (n > 0.0f) {
      l1 += (sNeg[c] / n) * A;          // per-class mean(-log pt) * Alpha
      const float mean = sPt[c] / n;
      float stdv = 0.0f;
      if (n > 1.0f) {
        const float var = (sPt2[c] - n * mean * mean) / (n - 1.0f); // ddof=1
        stdv = sqrtf(fmaxf(var, 0.0f));
      }
      l2raw += n * (stdv / mean) * e;   // loss2.mean() = sum_c n_c*loss2_c / N
    }
  }
  asum  = wave_sum(asum);
  l1    = wave_sum(l1);
  l2raw = wave_sum(l2raw);
  if (lane == 0)
    out[0] = l1 * (1.0f / (float)NUM_CLASSES) + (l2raw / asum) * invN;
}

__global__ void awb_zero(float* g) {
  const int i = blockIdx.x * blockDim.x + threadIdx.x;
  if (i < BINS) g[i] = 0.0f;
}

extern "C" void kernel_launch(void* const* d_in, const int* in_sizes, int n_in,
                              void* d_out, int out_size, void* d_ws, size_t ws_size,
                              hipStream_t stream) {
  const float* logits = (const float*)d_in[0];
  const int*   target = (const int*)d_in[1];
  float* out   = (float*)d_out;
  float* gbins = (float*)d_ws;

  const int N        = in_sizes[0] / NUM_CLASSES;   // 1,048,576 (divisible by TILE_ROWS)
  const int numTiles = N / TILE_ROWS;               // 4096

  const size_t shmem = 2 * (size_t)TILE_BYTES + BINS * sizeof(float); // 206,400 B
  (void)hipFuncSetAttribute((const void*)awb_main,
                            hipFuncAttributeMaxDynamicSharedMemorySize, (int)shmem);

  awb_zero<<<(BINS + 255) / 256, 256, 0, stream>>>(gbins);

  int grid = 512;                       // grid-stride; ~8 tiles per block
  if (grid > numTiles) grid = numTiles;
  awb_main<<<grid, BLOCK, shmem, stream>>>(logits, target, gbins, numTiles);

  awb_final<<<1, 32, 0, stream>>>(gbins, out, 1.0f / (float)N);
}